// Blast_56788057588465
// MI455X (gfx1250) — compile-verified
//
#include <hip/hip_runtime.h>

typedef __attribute__((ext_vector_type(2))) float v2f;
typedef __attribute__((ext_vector_type(8))) float v8f;

#define IN_DIM   4096
#define OUT_DIM  4096
#define BLK      256
#define RANK     16
#define NB       16      // blocks per dim
#define M_TOTAL  2048    // 2 * 1024 rows

// ---------------------------------------------------------------------------
// Stage 1: t[m, i*16+r] = sum_p x[m, i*256+p] * Vt[i, p, r]
// One wave computes a 16(m) x 16(r) tile for one (mtile, iblock), iterating
// K=256 in steps of 4 with V_WMMA_F32_16X16X4_F32 (exact f32 semantics).
// Fused: per-(row, iblock) partial row-sums of x (for the bias fold),
// accumulated from the A-fragments already in registers -> rs_part in ws.
// ---------------------------------------------------------------------------
__global__ __launch_bounds__(32) void k_stage1(const float* __restrict__ x,
                                               const float* __restrict__ Vt,
                                               float* __restrict__ t_ws,
                                               float* __restrict__ rs_part) {
    const int mtile = blockIdx.x;          // 0..127
    const int ib    = blockIdx.y;          // 0..15
    const int lane  = threadIdx.x;         // 0..31
    const int m     = lane & 15;
    const int khalf = lane >> 4;           // 0: K={0,1}, 1: K={2,3}
    const int k0    = khalf * 2;

    const float* xrow = x  + (size_t)(mtile * 16 + m) * IN_DIM + (size_t)ib * BLK;
    const float* vtb  = Vt + (size_t)ib * BLK * RANK;   // Vt[ib][p][r], row stride RANK

    v8f c = {};
    float psum = 0.0f;                     // partial row-sum of x over this chunk
    #pragma unroll 4
    for (int kbase = 0; kbase < BLK; kbase += 4) {
        // A fragment: x[m][kbase + k0], x[m][kbase + k0 + 1]  (8B-aligned b64 load)
        v2f a = *(const v2f*)(xrow + kbase + k0);
        // B fragment: Vt[kbase + k0][r=m], Vt[kbase + k0 + 1][r=m] (coalesced dwords)
        v2f b;
        b.x = vtb[(kbase + k0)     * RANK + m];
        b.y = vtb[(kbase + k0 + 1) * RANK + m];
        c = __builtin_amdgcn_wmma_f32_16x16x4_f32(false, a, false, b,
                                                  (short)0, c, false, false);
        psum += a.x + a.y;                 // co-executes with WMMA
    }

    // C/D layout: vgpr v -> row (v + 8*khalf), col = lane&15
    float* tp = t_ws + (size_t)(mtile * 16) * 256 + (size_t)ib * 16;
    #pragma unroll
    for (int v = 0; v < 8; ++v) {
        int mm = v + 8 * khalf;
        tp[(size_t)mm * 256 + m] = c[v];
    }

    // combine the two K-half lanes of row m; lanes 0..15 write the partial
    psum += __shfl_xor(psum, 16, 32);
    if (khalf == 0)
        rs_part[(size_t)(mtile * 16 + m) * NB + ib] = psum;
}

// ---------------------------------------------------------------------------
// Stage 2: u[o,m,r] = sum_i t[m,i,r]*S[o,i,r];  out = u @ U[o] + bias*(rs+1)
// One 256-thread workgroup per 16-row m-tile; t-tile, S, u and the row-sum
// reduction staged in LDS. Each of the 8 waves handles 2 o-blocks; per
// (o, n-tile): 4x V_WMMA_F32_16X16X4_F32 (K=16).
// ---------------------------------------------------------------------------
__global__ __launch_bounds__(256) void k_stage2(const float* __restrict__ t_ws,
                                                const float* __restrict__ rs_part,
                                                const float* __restrict__ S,
                                                const float* __restrict__ U,
                                                const float* __restrict__ bias,
                                                float* __restrict__ out) {
    __shared__ float sh_t[16 * 256];        // t[m][i*16+r]
    __shared__ float sh_S[16 * 256];        // S[o][i][r] -> o*256 + i*16 + r
    __shared__ float sh_u[16 * 16 * 16];    // u[o][m][r] -> o*256 + m*16 + r
    __shared__ float sh_red[256];           // row-sum partials (16 rows x 16 ib)
    __shared__ float sh_rs[16];             // reduced row sums

    const int mtile = blockIdx.x;           // 0..127
    const int tid   = threadIdx.x;
    const int lane  = tid & 31;
    const int wave  = tid >> 5;

    // Cooperative load: t-tile (16KB), all of S (16KB), row-sum partials (1KB)
    for (int idx = tid; idx < 16 * 256; idx += 256) {
        sh_t[idx] = t_ws[(size_t)mtile * (16 * 256) + idx];
        sh_S[idx] = S[idx];
    }
    sh_red[tid] = rs_part[(size_t)mtile * 256 + tid];   // row = tid>>4, ib = tid&15
    __syncthreads();

    // u: each thread owns one (m, r) pair, loops over all o (broadcast-friendly)
    {
        const int m = tid >> 4;
        const int r = tid & 15;
        float tv[16];
        #pragma unroll
        for (int i = 0; i < 16; ++i) tv[i] = sh_t[m * 256 + i * 16 + r];
        #pragma unroll
        for (int o = 0; o < 16; ++o) {
            float acc = 0.0f;
            #pragma unroll
            for (int i = 0; i < 16; ++i) acc += tv[i] * sh_S[o * 256 + i * 16 + r];
            sh_u[o * 256 + m * 16 + r] = acc;
        }
    }
    // reduce row-sum partials: 16 threads sum 16 values each
    if (tid < 16) {
        float s = 0.0f;
        #pragma unroll
        for (int j = 0; j < 16; ++j) s += sh_red[tid * 16 + j];
        sh_rs[tid] = s;
    }
    __syncthreads();

    const int m_lo  = lane & 15;
    const int khalf = lane >> 4;

    // bias scale per output row of the C fragment: (rowsum + 1)
    float rsf[8];
    #pragma unroll
    for (int v = 0; v < 8; ++v)
        rsf[v] = sh_rs[v + 8 * khalf] + 1.0f;

    for (int oo = 0; oo < 2; ++oo) {
        const int o = wave * 2 + oo;

        // A fragments for K=16 (4 wmma steps): u[o][m][ks*4 + 2*khalf + {0,1}]
        v2f afr[4];
        #pragma unroll
        for (int ks = 0; ks < 4; ++ks) {
            int k0 = ks * 4 + 2 * khalf;
            afr[ks].x = sh_u[o * 256 + m_lo * 16 + k0];
            afr[ks].y = sh_u[o * 256 + m_lo * 16 + k0 + 1];
        }

        const float* Ub = U + (size_t)o * RANK * BLK;  // U[o][r][q], row stride 256
        const float* bb = bias + o * BLK;

        for (int nt = 0; nt < 16; ++nt) {
            v8f c = {};
            #pragma unroll
            for (int ks = 0; ks < 4; ++ks) {
                int k0 = ks * 4 + 2 * khalf;
                v2f b;
                b.x = Ub[(size_t)(k0)     * BLK + nt * 16 + m_lo];
                b.y = Ub[(size_t)(k0 + 1) * BLK + nt * 16 + m_lo];
                c = __builtin_amdgcn_wmma_f32_16x16x4_f32(false, afr[ks], false, b,
                                                          (short)0, c, false, false);
            }
            const float bcol = bb[nt * 16 + m_lo];
            float* op = out + (size_t)(mtile * 16) * OUT_DIM + (size_t)o * BLK
                            + (size_t)nt * 16 + m_lo;
            #pragma unroll
            for (int v = 0; v < 8; ++v) {
                int mm = v + 8 * khalf;
                op[(size_t)mm * OUT_DIM] = c[v] + bcol * rsf[v];
            }
        }
    }
}

// ---------------------------------------------------------------------------
extern "C" void kernel_launch(void* const* d_in, const int* in_sizes, int n_in,
                              void* d_out, int out_size, void* d_ws, size_t ws_size,
                              hipStream_t stream) {
    (void)in_sizes; (void)n_in; (void)out_size; (void)ws_size;
    const float* x    = (const float*)d_in[0];   // (2,1024,4096)
    const float* S    = (const float*)d_in[1];   // (16,16,16)
    const float* U    = (const float*)d_in[2];   // (16,16,256)
    const float* Vt   = (const float*)d_in[3];   // (16,256,16)
    const float* bias = (const float*)d_in[4];   // (4096,)
    float* out  = (float*)d_out;                 // (2,1024,4096)

    float* t_ws    = (float*)d_ws;                    // 2048*256 floats = 2 MB
    float* rs_part = t_ws + (size_t)M_TOTAL * 256;    // 2048*16 floats = 128 KB

    dim3 g1(M_TOTAL / 16, NB);
    k_stage1<<<g1, 32, 0, stream>>>(x, Vt, t_ws, rs_part);
    k_stage2<<<M_TOTAL / 16, 256, 0, stream>>>(t_ws, rs_part, S, U, bias, out);
}